// SimAttn_89730456748848
// MI455X (gfx1250) — compile-verified
//
#include <hip/hip_runtime.h>

// Problem constants (match reference)
#define DIM   300
#define T1N   128
#define T2N   128
#define BATCH 4
#define ROWS  (BATCH * T1N)       // 512 rows for q (and for k/v: BATCH*T2N = 512)

#define WAVES 8
#define BLOCK (WAVES * 32)

typedef __attribute__((ext_vector_type(2))) float v2f;
typedef __attribute__((ext_vector_type(8))) float v8f;

// ---------------------------------------------------------------------------
// Projection: out[m,n] = sum_k X[m,k] * W[n,k] + bias[n]   (x @ W.T + b)
// One wave32 computes one 16x16 output tile with V_WMMA_F32_16X16X4_F32.
// A (16x4 f32) layout: lanes 0-15 hold M=lane, K={0,1}; lanes 16-31 K={2,3}.
// B (4x16 f32) layout mirrored: lanes 0-15 N=lane K={0,1}; lanes 16-31 K={2,3}.
// C/D (16x16 f32): VGPR r -> M = r + 8*(lane>=16), N = lane&15.
//
// Edge handling (N=300 is not a multiple of 16): B column n feeds only D
// column n, which is held by the same lane. So out-of-range lanes just load
// from a clamped-but-valid address (no mask, no divergence) and their WMMA
// results are simply never stored. All loads in the K-loop are unconditional
// coalesced b64s; two accumulators halve the WMMA dependency chain.
// ---------------------------------------------------------------------------
__global__ __launch_bounds__(32) void proj_wmma_kernel(
    const float* __restrict__ X, const float* __restrict__ W,
    const float* __restrict__ bias, float* __restrict__ out)
{
    const int lane  = threadIdx.x;      // 0..31, full wave, EXEC all-1s at WMMA
    const int nTile = blockIdx.x;       // 0..18  (ceil(300/16) = 19)
    const int mTile = blockIdx.y;       // 0..31  (512/16)
    const int m0 = mTile * 16;
    const int n0 = nTile * 16;

    const int half = lane >> 4;         // 0: K={0,1}   1: K={2,3}
    const int l16  = lane & 15;

    const int  n      = n0 + l16;
    const bool nvalid = (n < DIM);
    const int  nsafe  = nvalid ? n : (DIM - 1);   // clamped row, always valid

    const float* arow = X + (size_t)(m0 + l16) * DIM + 2 * half;
    const float* wrow = W + (size_t)nsafe      * DIM + 2 * half;

    v8f acc0 = {};
    v8f acc1 = {};
    // 300 = 37*8 + 4 : unroll-by-2 over K-steps of 4, plus one tail step.
    int kt = 0;
    for (; kt + 8 <= DIM; kt += 8) {
        v2f a0, b0, a1, b1;
        a0.x = arow[kt];     a0.y = arow[kt + 1];
        b0.x = wrow[kt];     b0.y = wrow[kt + 1];
        a1.x = arow[kt + 4]; a1.y = arow[kt + 5];
        b1.x = wrow[kt + 4]; b1.y = wrow[kt + 5];
        acc0 = __builtin_amdgcn_wmma_f32_16x16x4_f32(
            false, a0, false, b0, (short)0, acc0, false, false);
        acc1 = __builtin_amdgcn_wmma_f32_16x16x4_f32(
            false, a1, false, b1, (short)0, acc1, false, false);
    }
    {   // tail K-step (kt = 296)
        v2f a0, b0;
        a0.x = arow[kt]; a0.y = arow[kt + 1];
        b0.x = wrow[kt]; b0.y = wrow[kt + 1];
        acc0 = __builtin_amdgcn_wmma_f32_16x16x4_f32(
            false, a0, false, b0, (short)0, acc0, false, false);
    }
    const v8f acc = acc0 + acc1;

    if (nvalid) {
        const float bv = bias[n];
        #pragma unroll
        for (int r = 0; r < 8; ++r) {
            const int m = m0 + r + 8 * half;
            out[(size_t)m * DIM + n] = acc[r] + bv;
        }
    }
}

// ---------------------------------------------------------------------------
// Fused attention: one block per (b, i).
//   scores[j] = sum_d (q[b,i,d]+hL[b,i,j,d]) * (k[b,j,d]+hR[b,j,i,d])
//   p = softmax(scores); attn = clip(softmax(1000*p), 0, 1)
//   out[b,i,d] = sum_j attn[j] * (v[b,j,d] + hR[b,j,i,d])
// hL/hR rows are streamed once from HBM; the pass-2 hR re-read hits L2.
// ---------------------------------------------------------------------------
__global__ __launch_bounds__(BLOCK) void attn_fused_kernel(
    const float* __restrict__ q, const float* __restrict__ k,
    const float* __restrict__ v, const float* __restrict__ hL,
    const float* __restrict__ hR, float* __restrict__ out)
{
    const int b    = blockIdx.y;
    const int i    = blockIdx.x;
    const int tid  = threadIdx.x;
    const int wave = tid >> 5;
    const int lane = tid & 31;

    __shared__ float s_q[DIM];
    __shared__ float s_attn[T2N];

    const float* qrow = q + ((size_t)b * T1N + i) * DIM;
    for (int d = tid; d < DIM; d += BLOCK) s_q[d] = qrow[d];
    __syncthreads();

    // ---- Pass 1: scores, wave-per-j ----
    for (int j = wave; j < T2N; j += WAVES) {
        const float* hLp = hL + (((size_t)b * T1N + i) * T2N + j) * DIM;
        const float* hRp = hR + (((size_t)b * T2N + j) * T1N + i) * DIM;
        const float* kp  = k  + ((size_t)b * T2N + j) * DIM;
        float acc = 0.0f;
        for (int d = lane; d < DIM; d += 32)
            acc += (s_q[d] + hLp[d]) * (kp[d] + hRp[d]);
        #pragma unroll
        for (int off = 16; off > 0; off >>= 1)
            acc += __shfl_xor(acc, off, 32);
        if (lane == 0) s_attn[j] = acc;
    }
    __syncthreads();

    // ---- Softmax + sharpening (temp 1000) + clamp, on wave 0 ----
    if (wave == 0) {
        float m = -__builtin_inff();
        for (int j = lane; j < T2N; j += 32) m = fmaxf(m, s_attn[j]);
        #pragma unroll
        for (int off = 16; off > 0; off >>= 1) m = fmaxf(m, __shfl_xor(m, off, 32));

        float sum = 0.0f;
        for (int j = lane; j < T2N; j += 32) sum += expf(s_attn[j] - m);
        #pragma unroll
        for (int off = 16; off > 0; off >>= 1) sum += __shfl_xor(sum, off, 32);
        const float inv = 1.0f / sum;

        float m2 = -__builtin_inff();
        for (int j = lane; j < T2N; j += 32) {
            const float p = expf(s_attn[j] - m) * inv;
            m2 = fmaxf(m2, 1000.0f * p);
        }
        #pragma unroll
        for (int off = 16; off > 0; off >>= 1) m2 = fmaxf(m2, __shfl_xor(m2, off, 32));

        float sum2 = 0.0f;
        for (int j = lane; j < T2N; j += 32) {
            const float p = expf(s_attn[j] - m) * inv;
            sum2 += expf(1000.0f * p - m2);
        }
        #pragma unroll
        for (int off = 16; off > 0; off >>= 1) sum2 += __shfl_xor(sum2, off, 32);
        const float inv2 = 1.0f / sum2;

        for (int j = lane; j < T2N; j += 32) {
            const float p  = expf(s_attn[j] - m) * inv;
            const float a2 = expf(1000.0f * p - m2) * inv2;
            s_attn[j] = fminf(fmaxf(a2, 0.0f), 1.0f);
        }
    }
    __syncthreads();

    // ---- Pass 2: out[b,i,d] = sum_j attn[j]*(v[b,j,d] + hR[b,j,i,d]) ----
    for (int d = tid; d < DIM; d += BLOCK) {
        const float* vp  = v  + (size_t)b * T2N * DIM + d;
        const float* hRp = hR + ((size_t)b * T2N * T1N + i) * DIM + d;
        float acc = 0.0f;
        for (int j = 0; j < T2N; ++j)
            acc += s_attn[j] * (vp[(size_t)j * DIM] + hRp[(size_t)j * T1N * DIM]);
        out[((size_t)b * T1N + i) * DIM + d] = acc;
    }
}

extern "C" void kernel_launch(void* const* d_in, const int* in_sizes, int n_in,
                              void* d_out, int out_size, void* d_ws, size_t ws_size,
                              hipStream_t stream)
{
    const float* query = (const float*)d_in[0];
    const float* key   = (const float*)d_in[1];
    const float* value = (const float*)d_in[2];
    const float* hL    = (const float*)d_in[3];
    const float* hR    = (const float*)d_in[4];
    const float* WQ    = (const float*)d_in[5];
    const float* bQ    = (const float*)d_in[6];
    const float* WK    = (const float*)d_in[7];
    const float* bK    = (const float*)d_in[8];
    float* out = (float*)d_out;

    // Workspace: q, k, v projections (512 x 300 f32 each = 1.84 MB total)
    float* q = (float*)d_ws;
    float* k = q + (size_t)ROWS * DIM;
    float* v = k + (size_t)ROWS * DIM;

    const dim3 pgrid((DIM + 15) / 16, ROWS / 16);   // 19 x 32 tiles
    proj_wmma_kernel<<<pgrid, 32, 0, stream>>>(query, WQ, bQ, q);
    proj_wmma_kernel<<<pgrid, 32, 0, stream>>>(key,   WK, bK, k);
    proj_wmma_kernel<<<pgrid, 32, 0, stream>>>(value, WK, bK, v);  // v uses WK/bK (as in reference)

    attn_fused_kernel<<<dim3(T1N, BATCH), BLOCK, 0, stream>>>(q, k, v, hL, hR, out);
}